// CNN_Att_16887811408068
// MI455X (gfx1250) — compile-verified
//
#include <hip/hip_runtime.h>
#include <math.h>

// ---------------- constants from the reference ----------------
#define V_    50000
#define D_    100
#define WIN_  5
#define P_    411
#define LOCAL_ 100
#define NF_   100
#define GOUT_ 100
#define NCLS_ 2987
#define B_    256
#define LL_   415   // P + WIN - 1
#define LG_   411

typedef __attribute__((ext_vector_type(2))) float v2f;
typedef __attribute__((ext_vector_type(8))) float v8f;

static __device__ __forceinline__ v8f wmma_f32x4(v2f a, v2f b, v8f c) {
  // D = A(16x4) * B(4x16) + C(16x16), all f32.
  return __builtin_amdgcn_wmma_f32_16x16x4_f32(false, a, false, b, (short)0, c,
                                               false, false);
}

// ---------------- workspace layout (floats) ----------------
#define OF_ASUM    0                        // B*LL
#define OF_CSUM    (OF_ASUM + B_*LL_)       // B*LL
#define OF_COMP    (OF_CSUM + B_*LL_)       // 512
#define OF_POOLED  (OF_COMP + 512)          // B*300
#define OF_COMBINE (OF_POOLED + B_*300)     // B*200
#define OF_HIDDEN  (OF_COMBINE + B_*200)    // B*400

// ======================================================================
// K1: Asum[b,t] = sum_w tanh(E[b,t]·att_w[w] + att_b[w]); csum[b,t]=sum_d E
// ======================================================================
__global__ void k_att(const int* __restrict__ lidx, const float* __restrict__ emb,
                      const float* __restrict__ att_w, const float* __restrict__ att_b,
                      float* __restrict__ Asum, float* __restrict__ csum) {
  __shared__ float sw[WIN_ * D_];
  __shared__ float sb[WIN_];
  int tid = threadIdx.x;
  for (int i = tid; i < WIN_ * D_; i += 256) sw[i] = att_w[i];
  if (tid < WIN_) sb[tid] = att_b[tid];
  __syncthreads();
  int g = blockIdx.x * 256 + tid;
  if (g >= B_ * LL_) return;
  const float* e = emb + (long)lidx[g] * D_;
  float acc0 = 0.f, acc1 = 0.f, acc2 = 0.f, acc3 = 0.f, acc4 = 0.f, cs = 0.f;
  for (int d = 0; d < D_; ++d) {
    float ev = e[d];
    cs += ev;
    acc0 += ev * sw[0 * D_ + d];
    acc1 += ev * sw[1 * D_ + d];
    acc2 += ev * sw[2 * D_ + d];
    acc3 += ev * sw[3 * D_ + d];
    acc4 += ev * sw[4 * D_ + d];
  }
  float s = tanhf(acc0 + sb[0]) + tanhf(acc1 + sb[1]) + tanhf(acc2 + sb[2]) +
            tanhf(acc3 + sb[3]) + tanhf(acc4 + sb[4]);
  Asum[g] = s;
  csum[g] = cs;
}

// ======================================================================
// K2: comparison[p] = CR*min_{b,w} Asum[b,p+w] + CR*max_{b,w} Asum[b,p+w]
// ======================================================================
__global__ void k_comp(const float* __restrict__ Asum, float* __restrict__ comp) {
  int p = blockIdx.x;
  int tid = threadIdx.x;  // b
  const float* row = Asum + tid * LL_ + p;
  float mn = row[0], mx = row[0];
#pragma unroll
  for (int w = 1; w < WIN_; ++w) {
    float v = row[w];
    mn = fminf(mn, v);
    mx = fmaxf(mx, v);
  }
  __shared__ float smn[256], smx[256];
  smn[tid] = mn; smx[tid] = mx;
  __syncthreads();
  for (int s = 128; s > 0; s >>= 1) {
    if (tid < s) {
      smn[tid] = fminf(smn[tid], smn[tid + s]);
      smx[tid] = fmaxf(smx[tid], smx[tid + s]);
    }
    __syncthreads();
  }
  if (tid == 0) comp[p] = 0.8f * smn[0] + 0.8f * smx[0];
}

// ======================================================================
// K3: per-b judge -> sequential scan tw=(tw+x)/P -> local_units (combine[:,0:100])
// ======================================================================
__global__ void k_scan(const int* __restrict__ lidx, const float* __restrict__ emb,
                       const float* __restrict__ comp,
                       const float* __restrict__ att2_w, const float* __restrict__ att2_b,
                       const float* __restrict__ csum, float* __restrict__ combine) {
  __shared__ float sj[P_];
  __shared__ int   srow[P_];
  __shared__ float stw[D_];
  int b = blockIdx.x, tid = threadIdx.x;
  for (int p = tid; p < P_; p += 256) {
    sj[p]   = (csum[b * LL_ + p + 2] > comp[p]) ? 1.f : 0.f;
    srow[p] = lidx[b * LL_ + p + 2] * D_;
  }
  __syncthreads();
  if (tid < D_) {
    float tw = 0.f;
    for (int p = 0; p < P_; ++p) {
      float x = emb[srow[p] + tid] * sj[p];
      tw = (tw + x) / 411.0f;
    }
    stw[tid] = tw;
  }
  __syncthreads();
  if (tid < LOCAL_) {
    float lu = att2_b[tid];
    for (int d = 0; d < D_; ++d) lu += stw[d] * att2_w[tid * D_ + d];
    combine[b * 200 + tid] = lu;
  }
}

// ======================================================================
// K4: conv1/2/3 as one fused WMMA GEMM per (b, 16-filter tile) + ReLU + maxpool
//   Weight LDS layout (at LDS offset 0, all B loads fold into ds imm offsets):
//     sW[ s*1600 + k*64 + h*32 + f*2 + c ] = Wslice_s[f][4k + 2h + c]
//   -> per lane one ds_load_b64, half-waves hit disjoint bank halves.
//   2 t-tiles per wave: 12 WMMA per 6 LDS-b64 + 2 global-b64.
// ======================================================================
__global__ void __launch_bounds__(256)
k_conv(const int* __restrict__ gidx, const float* __restrict__ emb,
       const float* __restrict__ w1, const float* __restrict__ b1,
       const float* __restrict__ w2, const float* __restrict__ b2,
       const float* __restrict__ w3, const float* __restrict__ b3,
       float* __restrict__ pooled) {
  __shared__ float smem[9600 + 256 * 6 * 16];   // sW (38400B) then sS (98304B)
  float* sW = smem;                 // [6][25][2][16][2]
  float* sS = smem + 9600;          // [trow 0..255][s][f]
  int b = blockIdx.y;
  int ft = blockIdx.x;              // filter tile 0..6
  int tid = threadIdx.x;

  // stage all 6 weight slices for this filter tile into LDS (zero-pad f>=100)
  for (int i = tid; i < 9600; i += 256) {
    int c = i & 1;
    int f = (i >> 1) & 15;
    int h = (i >> 5) & 1;
    int rest = i >> 6;
    int k = rest % 25;
    int s = rest / 25;
    int d = 4 * k + 2 * h + c;
    int fg = ft * 16 + f;
    float v = 0.f;
    if (fg < NF_) {
      if (s == 0)      v = w1[fg * D_ + d];
      else if (s <= 2) v = w2[fg * 2 * D_ + (s - 1) * D_ + d];
      else             v = w3[fg * 3 * D_ + (s - 3) * D_ + d];
    }
    sW[i] = v;
  }
  __syncthreads();

  int lane = tid & 31, wave = tid >> 5;
  int halfsel = lane >> 4;          // 0: K pair {0,1}, 1: K pair {2,3}
  int lrow = lane & 15;
  const float* wbase = sW + halfsel * 32 + lrow * 2;   // + s*1600 + k*64 (imm)
  float pm1 = 0.f, pm2 = 0.f, pm3 = 0.f;               // relu>=0

  for (int phase = 0; phase < 2; ++phase) {
    int rowbase = phase * 160;      // phase0 rows 0..255, phase1 rows 160..415
    // this wave's two t-tiles (16 tiles per phase, 2 per wave; wave-uniform)
    int t0a = rowbase + (2 * wave) * 16;
    int t0b = t0a + 16;
    int ra = t0a + lrow; if (ra > LG_ - 1) ra = LG_ - 1;   // clamped rows never pooled
    int rb = t0b + lrow; if (rb > LG_ - 1) rb = LG_ - 1;
    const float* erow0 = emb + (long)gidx[b * LG_ + ra] * D_ + halfsel * 2;
    const float* erow1 = emb + (long)gidx[b * LG_ + rb] * D_ + halfsel * 2;
    __builtin_prefetch(erow0, 0, 1);
    __builtin_prefetch(erow1, 0, 1);
    v8f acc0[6] = {};
    v8f acc1[6] = {};
    for (int k = 0; k < 25; ++k) {
      v2f a0 = *(const v2f*)(erow0 + k * 4);
      v2f a1 = *(const v2f*)(erow1 + k * 4);
#pragma unroll
      for (int s = 0; s < 6; ++s) {
        v2f bb = *(const v2f*)(wbase + s * 1600 + k * 64);
        acc0[s] = wmma_f32x4(a0, bb, acc0[s]);
        acc1[s] = wmma_f32x4(a1, bb, acc1[s]);
      }
    }
#pragma unroll
    for (int s = 0; s < 6; ++s) {
#pragma unroll
      for (int v = 0; v < 8; ++v) {
        int ta = (2 * wave) * 16 + v + halfsel * 8;      // local rows 0..255
        int tb = ta + 16;
        sS[(ta * 6 + s) * 16 + lrow] = acc0[s][v];
        sS[(tb * 6 + s) * 16 + lrow] = acc1[s][v];
      }
    }
    __syncthreads();

    // pooling: 48 threads; cv = conv id, f = filter-in-tile
    int f = tid & 15, cv = tid >> 4;
    if (cv < 3) {
      int tlo = (phase == 0) ? 0 : 254;
      int thi = (phase == 0) ? 253 : (cv == 0 ? 410 : (cv == 1 ? 409 : 408));
      int fg = ft * 16 + f; if (fg > NF_ - 1) fg = NF_ - 1;
      float bias = (cv == 0 ? b1 : (cv == 1 ? b2 : b3))[fg];
      for (int t = tlo; t <= thi; ++t) {
        int tr = t - rowbase;
        float c;
        if (cv == 0)      c = sS[(tr * 6 + 0) * 16 + f];
        else if (cv == 1) c = sS[(tr * 6 + 1) * 16 + f] + sS[((tr + 1) * 6 + 2) * 16 + f];
        else              c = sS[(tr * 6 + 3) * 16 + f] + sS[((tr + 1) * 6 + 4) * 16 + f] +
                              sS[((tr + 2) * 6 + 5) * 16 + f];
        c = fmaxf(c + bias, 0.f);
        if (cv == 0) pm1 = fmaxf(pm1, c);
        else if (cv == 1) pm2 = fmaxf(pm2, c);
        else pm3 = fmaxf(pm3, c);
      }
    }
    __syncthreads();
  }

  int f = tid & 15, cv = tid >> 4;
  int fg = ft * 16 + f;
  if (cv < 3 && fg < NF_) {
    float pm = (cv == 0 ? pm1 : (cv == 1 ? pm2 : pm3));
    pooled[b * 300 + cv * NF_ + fg] = pm;
  }
}

// ======================================================================
// K5: global_units = pooled @ mf_w.T + mf_b  -> combine[:,100:200]
// ======================================================================
__global__ void k_mf(const float* __restrict__ pooled, const float* __restrict__ mf_w,
                     const float* __restrict__ mf_b, float* __restrict__ combine) {
  int g = blockIdx.x * 256 + threadIdx.x;
  if (g >= B_ * GOUT_) return;
  int b = g / GOUT_, o = g % GOUT_;
  float acc = mf_b[o];
  const float* pr = pooled + b * 300;
  const float* wr = mf_w + o * 300;
  for (int j = 0; j < 300; ++j) acc += pr[j] * wr[j];
  combine[b * 200 + 100 + o] = acc;
}

// ======================================================================
// K6: hidden = relu(combine @ fin_w.T + fin_b)
// ======================================================================
__global__ void k_fin(const float* __restrict__ combine, const float* __restrict__ fin_w,
                      const float* __restrict__ fin_b, float* __restrict__ hidden) {
  int g = blockIdx.x * 256 + threadIdx.x;
  if (g >= B_ * 400) return;
  int b = g / 400, j = g % 400;
  float acc = fin_b[j];
  const float* cr = combine + b * 200;
  const float* wr = fin_w + j * 200;
  for (int i = 0; i < 200; ++i) acc += cr[i] * wr[i];
  hidden[g] = fmaxf(acc, 0.f);
}

// ======================================================================
// K7: out = hidden @ fin2_w.T + fin2_b   (256x400 @ 400x2987) — f32 WMMA,
//     2 n-tiles per wave so the A tile is loaded once per 2 WMMA.
// ======================================================================
__global__ void __launch_bounds__(256)
k_fin2(const float* __restrict__ hidden, const float* __restrict__ w,
       const float* __restrict__ bias, float* __restrict__ out) {
  int tid = threadIdx.x;
  int lane = tid & 31, wave = tid >> 5;
  int nt0 = blockIdx.x * 16 + wave * 2;   // n tiles {nt0, nt0+1}, 187 valid
  int b0 = blockIdx.y * 16;
  int lrow = lane & 15, halfsel = lane >> 4;
  const float* arow = hidden + (b0 + lrow) * 400 + halfsel * 2;
  int ncol0 = nt0 * 16 + lrow;
  int ncol1 = ncol0 + 16;
  int cl0 = ncol0 > NCLS_ - 1 ? NCLS_ - 1 : ncol0;
  int cl1 = ncol1 > NCLS_ - 1 ? NCLS_ - 1 : ncol1;
  const float* brow0 = w + (long)cl0 * 400 + halfsel * 2;
  const float* brow1 = w + (long)cl1 * 400 + halfsel * 2;
  __builtin_prefetch(brow0, 0, 1);
  __builtin_prefetch(brow1, 0, 1);
  v8f acc0 = {};
  v8f acc1 = {};
  for (int k = 0; k < 100; ++k) {
    v2f a  = *(const v2f*)(arow + k * 4);
    v2f bb0 = *(const v2f*)(brow0 + k * 4);
    v2f bb1 = *(const v2f*)(brow1 + k * 4);
    acc0 = wmma_f32x4(a, bb0, acc0);
    acc1 = wmma_f32x4(a, bb1, acc1);
  }
  if (ncol0 < NCLS_) {
    float bs = bias[ncol0];
#pragma unroll
    for (int v = 0; v < 8; ++v)
      out[(b0 + v + halfsel * 8) * NCLS_ + ncol0] = acc0[v] + bs;
  }
  if (ncol1 < NCLS_) {
    float bs = bias[ncol1];
#pragma unroll
    for (int v = 0; v < 8; ++v)
      out[(b0 + v + halfsel * 8) * NCLS_ + ncol1] = acc1[v] + bs;
  }
}

// ======================================================================
extern "C" void kernel_launch(void* const* d_in, const int* in_sizes, int n_in,
                              void* d_out, int out_size, void* d_ws, size_t ws_size,
                              hipStream_t stream) {
  const int*   l_text  = (const int*)d_in[0];
  const int*   g_text  = (const int*)d_in[1];
  const float* emb     = (const float*)d_in[2];
  const float* att_w   = (const float*)d_in[3];
  const float* att_b   = (const float*)d_in[4];
  const float* att2_w  = (const float*)d_in[5];
  const float* att2_b  = (const float*)d_in[6];
  const float* conv1_w = (const float*)d_in[7];
  const float* conv1_b = (const float*)d_in[8];
  const float* conv2_w = (const float*)d_in[9];
  const float* conv2_b = (const float*)d_in[10];
  const float* conv3_w = (const float*)d_in[11];
  const float* conv3_b = (const float*)d_in[12];
  const float* mf_w    = (const float*)d_in[13];
  const float* mf_b    = (const float*)d_in[14];
  const float* fin_w   = (const float*)d_in[15];
  const float* fin_b   = (const float*)d_in[16];
  const float* fin2_w  = (const float*)d_in[17];
  const float* fin2_b  = (const float*)d_in[18];
  float* out = (float*)d_out;
  float* ws  = (float*)d_ws;

  float* Asum    = ws + OF_ASUM;
  float* csum    = ws + OF_CSUM;
  float* comp    = ws + OF_COMP;
  float* pooled  = ws + OF_POOLED;
  float* combine = ws + OF_COMBINE;
  float* hidden  = ws + OF_HIDDEN;

  k_att <<<(B_ * LL_ + 255) / 256, 256, 0, stream>>>(l_text, emb, att_w, att_b, Asum, csum);
  k_comp<<<P_, 256, 0, stream>>>(Asum, comp);
  k_scan<<<B_, 256, 0, stream>>>(l_text, emb, comp, att2_w, att2_b, csum, combine);
  k_conv<<<dim3(7, B_), 256, 0, stream>>>(g_text, emb, conv1_w, conv1_b, conv2_w,
                                          conv2_b, conv3_w, conv3_b, pooled);
  k_mf  <<<(B_ * GOUT_ + 255) / 256, 256, 0, stream>>>(pooled, mf_w, mf_b, combine);
  k_fin <<<(B_ * 400 + 255) / 256, 256, 0, stream>>>(combine, fin_w, fin_b, hidden);
  k_fin2<<<dim3(12, B_ / 16), 256, 0, stream>>>(hidden, fin2_w, fin2_b, out);
}